// GaussianModel_32280974197216
// MI455X (gfx1250) — compile-verified
//
#include <hip/hip_runtime.h>

#define EPS_EYE 1e-7f

// ---- gfx1250 async LDS-DMA path, guarded so the file always compiles ----
#if defined(__has_builtin)
#  if __has_builtin(__builtin_amdgcn_global_load_async_to_lds_b128)
#    define HAVE_ASYNC_LOAD 1
#  endif
#  if __has_builtin(__builtin_amdgcn_global_store_async_from_lds_b128)
#    define HAVE_ASYNC_STORE 1
#  endif
#  if __has_builtin(__builtin_amdgcn_s_wait_asynccnt)
#    define WAIT_ASYNC() __builtin_amdgcn_s_wait_asynccnt(0)
#  endif
#endif
#ifndef HAVE_ASYNC_LOAD
#  define HAVE_ASYNC_LOAD 0
#endif
#ifndef HAVE_ASYNC_STORE
#  define HAVE_ASYNC_STORE 0
#endif
#ifndef WAIT_ASYNC
#  define WAIT_ASYNC() asm volatile("s_wait_asynccnt 0" ::: "memory")
#endif

// Builtins take typed int4 pointers: AS(1) for global, AS(3) for LDS.
typedef int v4i __attribute__((ext_vector_type(4)));
typedef __attribute__((address_space(1))) v4i g_v4i;
typedef __attribute__((address_space(3))) v4i l_v4i;

__global__ __launch_bounds__(256)
void cov3d_kernel(const float* __restrict__ scaling,
                  const float* __restrict__ rotation,
                  float* __restrict__ out, int N)
{
    __shared__ float sIn[256 * 3];   // staged scaling, 3 KB
    __shared__ float sOut[256 * 9];  // staged covariance, 9 KB

    const int t = threadIdx.x;
    const long long base = (long long)blockIdx.x * 256;
    long long rem = (long long)N - base;
    const int nElem = rem > 256 ? 256 : (int)rem;

    // ---------- stage scaling (stride-3 AoS) into LDS, contiguous b128 ----------
    const float* gsc = scaling + base * 3;   // 16B-aligned: 256*3*4 = 3072 B per block
    const int F = nElem * 3;
    const int full4 = F >> 2;
#if HAVE_ASYNC_LOAD
    if (t < full4) {
        __builtin_amdgcn_global_load_async_to_lds_b128(
            (g_v4i*)(gsc + 4 * t),
            (l_v4i*)(sIn + 4 * t),
            0, 0);
    }
    WAIT_ASYNC();
#else
    if (t < full4) {
        ((float4*)sIn)[t] = ((const float4*)gsc)[t];
    }
#endif
    if ((F & 3) && t == full4) {             // tail floats (never OOB for this layout)
        for (int r = full4 * 4; r < F; ++r) sIn[r] = gsc[r];
    }
    __syncthreads();

    // ---------- per-Gaussian register math ----------
    if (t < nElem) {
        const long long i = base + t;
        const float4 q = ((const float4*)rotation)[i];   // coalesced 16 B/lane

        const float n2  = q.x * q.x + q.y * q.y + q.z * q.z + q.w * q.w;
        const float inv = 1.0f / sqrtf(n2);
        const float x = q.x * inv, y = q.y * inv, z = q.z * inv, w = q.w * inv;

        // LDS reads: lane stride 3 floats, gcd(3,64)=1 -> conflict-free
        const float s0 = expf(sIn[3 * t + 0]);
        const float s1 = expf(sIn[3 * t + 1]);
        const float s2 = expf(sIn[3 * t + 2]);

        const float l00 = (1.0f - 2.0f * (y * y + z * z)) * s0;
        const float l01 = (2.0f * (x * y - w * z)) * s1;
        const float l02 = (2.0f * (x * z + w * y)) * s2;
        const float l10 = (2.0f * (x * y + w * z)) * s0;
        const float l11 = (1.0f - 2.0f * (x * x + z * z)) * s1;
        const float l12 = (2.0f * (y * z - w * x)) * s2;
        const float l20 = (2.0f * (x * z - w * y)) * s0;
        const float l21 = (2.0f * (y * z + w * x)) * s1;
        const float l22 = (1.0f - 2.0f * (x * x + y * y)) * s2;

        const float c00 = l00 * l00 + l01 * l01 + l02 * l02;
        const float c01 = l00 * l10 + l01 * l11 + l02 * l12;
        const float c02 = l00 * l20 + l01 * l21 + l02 * l22;
        const float c11 = l10 * l10 + l11 * l11 + l12 * l12;
        const float c12 = l10 * l20 + l11 * l21 + l12 * l22;
        const float c22 = l20 * l20 + l21 * l21 + l22 * l22;

        // LDS writes: lane stride 9 floats, gcd(9,64)=1 -> conflict-free
        float* o = sOut + t * 9;
        o[0] = c00 + EPS_EYE; o[1] = c01;           o[2] = c02;
        o[3] = c01;           o[4] = c11 + EPS_EYE; o[5] = c12;
        o[6] = c02;           o[7] = c12;           o[8] = c22 + EPS_EYE;
    }
    __syncthreads();

    // ---------- stream out: contiguous b128 from LDS ----------
    const int G = nElem * 9;
    float* gout = out + base * 9;            // 16B-aligned: 256*9*4 = 9216 B per block
    const int gfull4 = G >> 2;
#if HAVE_ASYNC_STORE
    for (int v = t; v < gfull4; v += 256) {
        __builtin_amdgcn_global_store_async_from_lds_b128(
            (g_v4i*)(gout + 4 * v),
            (l_v4i*)(sOut + 4 * v),
            0, 0);
    }
    if ((G & 3) && t == 0) {
        for (int r = gfull4 * 4; r < G; ++r) gout[r] = sOut[r];
    }
    // trailing ASYNCcnt is drained by S_ENDPGM's implicit wait-idle
#else
    for (int v = t; v < gfull4; v += 256) {
        ((float4*)gout)[v] = ((const float4*)sOut)[v];
    }
    if ((G & 3) && t == 0) {
        for (int r = gfull4 * 4; r < G; ++r) gout[r] = sOut[r];
    }
#endif
}

extern "C" void kernel_launch(void* const* d_in, const int* in_sizes, int n_in,
                              void* d_out, int out_size, void* d_ws, size_t ws_size,
                              hipStream_t stream)
{
    (void)n_in; (void)out_size; (void)d_ws; (void)ws_size;
    const float* scaling  = (const float*)d_in[0];
    const float* rotation = (const float*)d_in[1];
    float* out = (float*)d_out;

    const int N = in_sizes[0] / 3;           // scaling is (N,3)
    const int blocks = (N + 255) / 256;
    hipLaunchKernelGGL(cov3d_kernel, dim3(blocks), dim3(256), 0, stream,
                       scaling, rotation, out, N);
}